// RqQuantizer_29180007809310
// MI455X (gfx1250) — compile-verified
//
#include <hip/hip_runtime.h>

typedef float v2f __attribute__((ext_vector_type(2)));
typedef float v8f __attribute__((ext_vector_type(8)));

#define DIMD  256
#define NEMB  1024
#define NTOK  32768
#define RQ_DEPTH 4
#define BETAF 10.0f
#define BPAD  260   // 256 + 4 pad floats -> conflict-free ds_load_b64

// ---------------------------------------------------------------------------
// Fused GEMM (f32 WMMA) + argmin over K=1024 codes.
// Block = 256 threads = 8 waves; each wave owns 16 rows (A in registers),
// block stages one 16-code emb tile in LDS per column-tile iteration.
// ---------------------------------------------------------------------------
__global__ __launch_bounds__(256)
void rq_score_argmin(const float* __restrict__ res,
                     const float* __restrict__ emb,
                     const float* __restrict__ e2,
                     int* __restrict__ codes, int depth)
{
    __shared__ float lds_b[16 * BPAD];

    const int tid  = threadIdx.x;
    const int wave = tid >> 5;
    const int lane = tid & 31;
    const int half = lane >> 4;       // 0: K0/K1 lanes, 1: K2/K3 lanes
    const int l    = lane & 15;
    const int row0 = blockIdx.x * 128 + wave * 16;

    // A-tile (16 rows x 256) in registers: per lane, row = row0+l,
    // v2f i holds K = 4*i + 2*half + {0,1}  (WMMA f32 16x16x4 A layout)
    v2f a[64];
    const float* aptr = res + (size_t)(row0 + l) * DIMD + 2 * half;
#pragma unroll
    for (int i = 0; i < 64; ++i)
        a[i] = *(const v2f*)(aptr + i * 4);

    float minv[8];
    int   mini[8];
#pragma unroll
    for (int r = 0; r < 8; ++r) { minv[r] = 3.0e38f; mini[r] = 0; }

    for (int t = 0; t < 64; ++t) {
        // Cooperative stage of emb tile [t*16 .. t*16+16) x 256 into LDS.
        // 4096 floats / 256 threads = 4 x float4 each; padded rows stay 16B aligned.
        {
            const float4* src = (const float4*)(emb + (size_t)t * 16 * DIMD);
#pragma unroll
            for (int k = 0; k < 4; ++k) {
                int f  = tid + k * 256;       // float4 index in tile (0..1023)
                int rr = f >> 6;              // code row 0..15
                int cc = (f & 63) << 2;       // column within row
                *(float4*)(&lds_b[rr * BPAD + cc]) = src[f];
            }
        }
        __syncthreads();

        v8f acc = {0.f, 0.f, 0.f, 0.f, 0.f, 0.f, 0.f, 0.f};
        // B layout mirrors A: lane -> code column l, v2f i -> K = 4*i+2*half+{0,1}
        const float* bbase = &lds_b[l * BPAD + 2 * half];
#pragma unroll
        for (int i = 0; i < 64; ++i) {
            v2f b = *(const v2f*)(bbase + i * 4);
            acc = __builtin_amdgcn_wmma_f32_16x16x4_f32(
                false, a[i], false, b, (short)0, acc, false, false);
        }

        // score = ||e||^2 - 2*dot ; element (vgpr r, lane) covers
        // row = half*8 + r (of this wave's tile), col = t*16 + l
        float e2v = e2[t * 16 + l];
        int   col = t * 16 + l;
#pragma unroll
        for (int r = 0; r < 8; ++r) {
            float s = fmaf(-2.0f, acc[r], e2v);
            if (s < minv[r]) { minv[r] = s; mini[r] = col; }
        }
        __syncthreads();   // before overwriting lds_b next iteration
    }

    // Argmin reduction across the 16 lanes of each half (tie -> lowest index,
    // matching jnp.argmin). Lane l==0 of each half holds the result.
#pragma unroll
    for (int r = 0; r < 8; ++r) {
#pragma unroll
        for (int off = 8; off > 0; off >>= 1) {
            float ov = __shfl_xor(minv[r], off, 16);
            int   oi = __shfl_xor(mini[r], off, 16);
            if (ov < minv[r] || (ov == minv[r] && oi < mini[r])) {
                minv[r] = ov; mini[r] = oi;
            }
        }
        if (l == 0) {
            int row = row0 + half * 8 + r;
            codes[(size_t)row * RQ_DEPTH + depth] = mini[r];
        }
    }
}

// ---------------------------------------------------------------------------
// res -= emb[code]; accumulate sum(res^2) deterministically per block.
// One thread per float4 of res.
// ---------------------------------------------------------------------------
__global__ __launch_bounds__(256)
void rq_update(float* __restrict__ res, const float* __restrict__ emb,
               const int* __restrict__ codes, int depth,
               float* __restrict__ partial)
{
    __shared__ float red[256];
    int idx = blockIdx.x * 256 + threadIdx.x;   // float4 index
    int n   = idx >> 6;
    int j   = (idx & 63) << 2;
    int c   = codes[(size_t)n * RQ_DEPTH + depth];

    float4  q  = *(const float4*)(emb + (size_t)c * DIMD + j);
    float4* rp = (float4*)(res + (size_t)n * DIMD + j);
    float4  r  = *rp;
    r.x -= q.x; r.y -= q.y; r.z -= q.z; r.w -= q.w;
    *rp = r;

    red[threadIdx.x] = r.x * r.x + r.y * r.y + r.z * r.z + r.w * r.w;
    __syncthreads();
#pragma unroll
    for (int o = 128; o > 0; o >>= 1) {
        if (threadIdx.x < o) red[threadIdx.x] += red[threadIdx.x + o];
        __syncthreads();
    }
    if (threadIdx.x == 0) partial[blockIdx.x] += red[0];
}

// ---------------------------------------------------------------------------
// res = transpose(z) : [B,C,H,W] -> [B*H*W, C]; also zero loss partials.
// ---------------------------------------------------------------------------
__global__ __launch_bounds__(256)
void rq_init(const float* __restrict__ z, float* __restrict__ res,
             float* __restrict__ partial)
{
    int i  = blockIdx.x * 256 + threadIdx.x;  // res flat index (coalesced write)
    int n  = i >> 8;
    int c  = i & 255;
    int b  = n >> 10;
    int hw = n & 1023;
    res[i] = z[((size_t)b * DIMD + c) * 1024 + hw];
    if (i < 8192) partial[i] = 0.0f;
}

// ---------------------------------------------------------------------------
// e2[k] = ||emb[k]||^2
// ---------------------------------------------------------------------------
__global__ __launch_bounds__(256)
void rq_e2(const float* __restrict__ emb, float* __restrict__ e2)
{
    int k = blockIdx.x * 256 + threadIdx.x;
    const float4* p = (const float4*)(emb + (size_t)k * DIMD);
    float s = 0.0f;
#pragma unroll 8
    for (int i = 0; i < 64; ++i) {
        float4 v = p[i];
        s += v.x * v.x + v.y * v.y + v.z * v.z + v.w * v.w;
    }
    e2[k] = s;
}

// ---------------------------------------------------------------------------
// z_q[b,c,h,w] = z - res_final  (== agg, the straight-through output)
// ---------------------------------------------------------------------------
__global__ __launch_bounds__(256)
void rq_zq(const float* __restrict__ z, const float* __restrict__ res,
           float* __restrict__ zq)
{
    int i  = blockIdx.x * 256 + threadIdx.x;  // z flat index (coalesced)
    int hw = i & 1023;
    int c  = (i >> 10) & 255;
    int b  = i >> 18;
    int n  = b * 1024 + hw;
    zq[i] = z[i] - res[(size_t)n * DIMD + c];
}

// ---------------------------------------------------------------------------
// loss = (1+beta)/(N*D) * sum(partials)   (fixed-order, deterministic)
// ---------------------------------------------------------------------------
__global__ __launch_bounds__(256)
void rq_loss(const float* __restrict__ partial, float* __restrict__ loss)
{
    __shared__ float red[256];
    float s = 0.0f;
    for (int k = threadIdx.x; k < 8192; k += 256) s += partial[k];
    red[threadIdx.x] = s;
    __syncthreads();
#pragma unroll
    for (int o = 128; o > 0; o >>= 1) {
        if (threadIdx.x < o) red[threadIdx.x] += red[threadIdx.x + o];
        __syncthreads();
    }
    if (threadIdx.x == 0)
        *loss = red[0] * (1.0f + BETAF) / (float)((size_t)NTOK * DIMD);
}

// ---------------------------------------------------------------------------
extern "C" void kernel_launch(void* const* d_in, const int* in_sizes, int n_in,
                              void* d_out, int out_size, void* d_ws, size_t ws_size,
                              hipStream_t stream)
{
    const float* z   = (const float*)d_in[0];   // [32,256,32,32] f32
    const float* emb = (const float*)d_in[1];   // [1024,256] f32

    float* out   = (float*)d_out;
    float* zq    = out;                         // 8388608 f32
    float* loss  = out + 8388608;               // 1 f32
    int*   codes = (int*)(out + 8388609);       // 131072 i32 [N, DEPTH]

    char*  ws      = (char*)d_ws;
    float* res     = (float*)ws;                        // 32 MB
    float* e2      = (float*)(ws + 33554432);           // 4 KB
    float* partial = (float*)(ws + 33554432 + 4096);    // 32 KB

    rq_init<<<NTOK * DIMD / 256, 256, 0, stream>>>(z, res, partial);
    rq_e2<<<NEMB / 256, 256, 0, stream>>>(emb, e2);

    for (int d = 0; d < RQ_DEPTH; ++d) {
        rq_score_argmin<<<NTOK / 128, 256, 0, stream>>>(res, emb, e2, codes, d);
        rq_update<<<NTOK * DIMD / 4 / 256, 256, 0, stream>>>(res, emb, codes, d, partial);
    }

    rq_zq<<<NTOK * DIMD / 256, 256, 0, stream>>>(z, res, zq);
    rq_loss<<<1, 256, 0, stream>>>(partial, loss);
}